// AssigmentLayer_90074054132588
// MI455X (gfx1250) — compile-verified
//
#include <hip/hip_runtime.h>
#include <cstdint>

// ---------------------------------------------------------------------------
// out[b,j,2i+c] = inputs[m/30,0,c] * w_c[m%30], m = j*B + b - i (0 if m<0)
// out[b,j,20+u] = inputs[b,j,2+u]
// Memory-bound: ~600MB traffic -> ~26us @ 23.3TB/s. No WMMA (0.2 GFLOP total).
// CDNA5 paths used: global_load_async_to_lds (w tables), s_wait_asynccnt,
// non-temporal TH_NT loads/stores for the streamed regions.
// ---------------------------------------------------------------------------

typedef float v2f __attribute__((ext_vector_type(2)));
typedef float v4f __attribute__((ext_vector_type(4)));

typedef __attribute__((address_space(1))) int* gas_i32p;  // global
typedef __attribute__((address_space(3))) int* las_i32p;  // LDS

#if defined(__gfx1250__) && __has_builtin(__builtin_amdgcn_global_load_async_to_lds_b32)
#define HAVE_ASYNC_LDS 1
#else
#define HAVE_ASYNC_LDS 0
#endif

static __device__ __forceinline__ void wait_async_then_barrier() {
#if HAVE_ASYNC_LDS
#if __has_builtin(__builtin_amdgcn_s_wait_asynccnt)
    __builtin_amdgcn_s_wait_asynccnt(0);
#else
    asm volatile("s_wait_asynccnt 0" ::: "memory");
#endif
#endif
    __syncthreads();
}

// Products region: 5 float4 per row (channels 0..19).
__global__ __launch_bounds__(256) void assign_weights_kernel(
    const float* __restrict__ in, const float* __restrict__ w1,
    const float* __restrict__ w2, float* __restrict__ out,
    unsigned B, unsigned total) {
    // Interleaved weight table in LDS: wl[2t] = w1[t], wl[2t+1] = w2[t].
    __shared__ float wl[64];
    unsigned tid = threadIdx.x;
#if HAVE_ASYNC_LDS
    if (tid < 60u) {
        const float* src = (tid < 30u) ? (w1 + tid) : (w2 + (tid - 30u));
        unsigned loff = (tid < 30u) ? (tid * 2u) : ((tid - 30u) * 2u + 1u);
        __builtin_amdgcn_global_load_async_to_lds_b32(
            (gas_i32p)src, (las_i32p)&wl[loff], 0, 0);
    }
#else
    if (tid < 60u) {
        unsigned t = (tid < 30u) ? tid : (tid - 30u);
        wl[(tid < 30u) ? (t * 2u) : (t * 2u + 1u)] =
            (tid < 30u) ? w1[t] : w2[t];
    }
#endif
    wait_async_then_barrier();

    unsigned idx = blockIdx.x * 256u + tid;
    if (idx >= total) return;

    unsigned r = idx / 5u;           // row = b*30 + j
    unsigned q = idx - r * 5u;       // which float4 (channels 4q..4q+3)
    unsigned b = r / 30u;
    unsigned j = r - b * 30u;

    unsigned m0  = j * B + b;        // a = j*B + b
    unsigned bhi = m0 / 30u;
    unsigned t0  = m0 - bhi * 30u;

    v4f o;
    unsigned i0 = q * 2u;
#pragma unroll
    for (int s = 0; s < 2; ++s) {
        unsigned i = i0 + (unsigned)s;
        float v0 = 0.0f, v1 = 0.0f;
        int bb;
        unsigned t;
        if (i <= t0) { bb = (int)bhi;      t = t0 - i; }
        else         { bb = (int)bhi - 1;  t = t0 + 30u - i; }
        if (bb >= 0) {
            // 8B-aligned cached load of inputs[bb,0,0:2] (reused via L2).
            v2f sv = *(const v2f*)(in + (unsigned)bb * 300u);
            v2f wv = *(const v2f*)&wl[t * 2u];   // ds_load_b64
            v0 = sv.x * wv.x;
            v1 = sv.y * wv.y;
        }
        o[2 * s]     = v0;
        o[2 * s + 1] = v1;
    }
    // Streaming 440MB output: non-temporal, 16B aligned.
    __builtin_nontemporal_store(o, (v4f*)out + (r * 7u + q));
}

// Copy region: channels 20..27 = inputs[b,j,2..9]; 2 float4 per row.
__global__ __launch_bounds__(256) void assign_copy_kernel(
    const float* __restrict__ in, float* __restrict__ out, unsigned total) {
    unsigned idx = blockIdx.x * 256u + threadIdx.x;
    if (idx >= total) return;
    unsigned r = idx >> 1;           // row = b*30 + j
    unsigned u = idx & 1u;           // which half of the 8 copied floats
    unsigned e = r * 10u + 2u + u * 4u;   // inputs flat index (8B aligned)
    v2f a = __builtin_nontemporal_load((const v2f*)(in + e));
    v2f c = __builtin_nontemporal_load((const v2f*)(in + e + 2u));
    v4f o;
    o.x = a.x; o.y = a.y; o.z = c.x; o.w = c.y;
    __builtin_nontemporal_store(o, (v4f*)out + (r * 7u + 5u + u));
}

extern "C" void kernel_launch(void* const* d_in, const int* in_sizes, int n_in,
                              void* d_out, int out_size, void* d_ws, size_t ws_size,
                              hipStream_t stream) {
    const float* in = (const float*)d_in[0];
    const float* w1 = (const float*)d_in[1];
    const float* w2 = (const float*)d_in[2];
    float* out = (float*)d_out;

    const unsigned B = (unsigned)(in_sizes[0] / 300);  // B*30*10 elements
    const unsigned rows = B * 30u;
    const unsigned totalA = rows * 5u;  // 5 float4 per row (weight region)
    const unsigned totalB = rows * 2u;  // 2 float4 per row (copy region)

    assign_weights_kernel<<<(totalA + 255u) / 256u, 256, 0, stream>>>(
        in, w1, w2, out, B, totalA);
    assign_copy_kernel<<<(totalB + 255u) / 256u, 256, 0, stream>>>(
        in, out, totalB);
}